// Defog_32014686224842
// MI455X (gfx1250) — compile-verified
//
#include <hip/hip_runtime.h>

// Problem constants (match reference setup_inputs)
#define B_    8
#define C_    3
#define H_    768
#define W_    1024
#define HW_   (H_ * W_)
#define N_    (B_ * HW_)        // 6,291,456 pixels per plane
#define RG    81                // guided filter radius
#define KK2   26569.0f          // 163*163
#define RM    7                 // min-filter radius
#define BINS_ 2000
#define EPSG  1e-3f
#define SEG   96                // vertical-pass segment height
#define NSEG  (H_ / SEG)        // 8
#define FBBLK 96                // per-image partial-sum blocks

#define LDS_AS __attribute__((address_space(3)))

typedef __attribute__((ext_vector_type(2))) float v2f;
typedef __attribute__((ext_vector_type(8))) float v8f;

#if defined(__has_builtin)
#if __has_builtin(__builtin_amdgcn_wmma_f32_16x16x4_f32)
#define HAVE_WMMA_F32X4 1
#endif
#endif

// ---- CDNA5 async global->LDS (ASYNCcnt-tracked DMA; bypasses VGPRs) ----
__device__ __forceinline__ void async_g2l_b128(LDS_AS float* lp, const float* gp) {
    asm volatile("global_load_async_to_lds_b128 %0, %1, off" :: "v"(lp), "v"(gp) : "memory");
}
__device__ __forceinline__ void wait_async0() {
    asm volatile("s_wait_asynccnt 0x0" ::: "memory");
}

// ---- ordered-float <-> uint encoding for atomic min/max ----
__device__ __forceinline__ unsigned fenc(float f) {
    unsigned u = __float_as_uint(f);
    return (u & 0x80000000u) ? ~u : (u | 0x80000000u);
}
__device__ __forceinline__ float fdec(unsigned u) {
    return __uint_as_float((u & 0x80000000u) ? (u ^ 0x80000000u) : ~u);
}

// ---- inclusive scan of 256 f32 partials ----
// WMMA version: X[r,c] = part[16c+r]; Y = L(16x16 lower-tri ones) * X via a
// chain of 4x V_WMMA_F32_16X16X4_F32 (K-chunks of 4). Column totals from
// Y[15][c] (acc[7], lanes 16..31 per the documented C/D layout); cross-column
// offsets added per lane; scatter S back to LDS. Executed by wave 0 only so
// EXEC is all-ones inside the WMMA region (ISA requirement).
__device__ __forceinline__ void scan256_wave0(float* part, float* tt) {
#ifdef HAVE_WMMA_F32X4
    const int l = threadIdx.x;       // 0..31
    const int half = l >> 4;         // 0: rows 0..7 / K {4q,4q+1};  1: rows 8..15 / K {4q+2,4q+3}
    const int lc = l & 15;
    v8f acc = {};
    for (int q = 0; q < 4; ++q) {
        const int k0 = 4 * q + 2 * half;
        v2f a, b;
        a.x = (lc >= k0) ? 1.0f : 0.0f;       // A[m][k0],   m = lc (A 16x4 layout)
        a.y = (lc >= k0 + 1) ? 1.0f : 0.0f;   // A[m][k0+1]
        b.x = part[16 * lc + k0];             // B[k0][c],   c = lc (B 4x16 layout)
        b.y = part[16 * lc + k0 + 1];         // B[k0+1][c]
        acc = __builtin_amdgcn_wmma_f32_16x16x4_f32(false, a, false, b, (short)0,
                                                    acc, false, false);
    }
    if (half) tt[lc] = acc[7];                // t_c = Y[15][c]
    asm volatile("s_wait_dscnt 0x0" ::: "memory");
    float off = 0.0f;
    for (int c2 = 0; c2 < 15; ++c2)
        if (c2 < lc) off += tt[c2];           // off_c = sum_{c'<c} t_c'
    const int mbase = half * 8;
    for (int j = 0; j < 8; ++j)
        part[16 * lc + mbase + j] = acc[j] + off;
#else
    (void)part; (void)tt;
#endif
}

// Hillis-Steele fallback (whole block). Only compiled in if no WMMA builtin.
__device__ __forceinline__ void scan256_fallback(float* part, int t) {
    for (int off = 1; off < 256; off <<= 1) {
        float add = (t >= off) ? part[t - off] : 0.0f;
        __syncthreads();
        part[t] += add;
        __syncthreads();
    }
}

__device__ __forceinline__ void scan256(float* part, float* tt, int t) {
#ifdef HAVE_WMMA_F32X4
    __syncthreads();
    if (t < 32) scan256_wave0(part, tt);
    __syncthreads();
#else
    __syncthreads();
    scan256_fallback(part, t);
#endif
}

// aux layout (uint words):
// [0] mn_enc  [1] mx_enc  [2] thresh(f32)  [3] A(f32)
// [4] anyflag [5] Acand_enc [6] fallback(f32)
// [16 .. 16+2000) histogram counts
#define AUX_WORDS 2112

// ---------------- init ----------------
__global__ __launch_bounds__(256) void k_init(unsigned* aux) {
    for (int i = threadIdx.x; i < AUX_WORDS; i += 256) aux[i] = 0u;
    __syncthreads();
    if (threadIdx.x == 0) {
        aux[0] = 0xFFFFFFFFu;  // running min (encoded)
        aux[1] = 0u;           // running max (encoded)
    }
}

// ---------------- dark channel: V1 = 255 * min_c x ----------------
__global__ __launch_bounds__(256) void k_dark(const float* __restrict__ x,
                                              float* __restrict__ v1) {
    int i = blockIdx.x * 256 + threadIdx.x;
    if (i >= N_) return;
    int b = i / HW_, r = i - b * HW_;
    const float* p = x + (size_t)b * 3 * HW_ + r;
    float v = fminf(p[0], fminf(p[HW_], p[2 * HW_]));
    v1[i] = v * 255.0f;
}

// ---------------- 15x15 min filter (separable, +inf padding) ----------------
__global__ __launch_bounds__(256) void k_rowmin(const float* __restrict__ in,
                                                float* __restrict__ out) {
    int i = blockIdx.x * 256 + threadIdx.x;
    if (i >= N_) return;
    int xx = i % W_;
    int x0 = max(xx - RM, 0), x1 = min(xx + RM, W_ - 1);
    const float* row = in + (i - xx);
    float v = 3.0e38f;
    for (int xc = x0; xc <= x1; ++xc) v = fminf(v, row[xc]);
    out[i] = v;
}

__global__ __launch_bounds__(256) void k_colmin(const float* __restrict__ in,
                                                float* __restrict__ out) {
    int i = blockIdx.x * 256 + threadIdx.x;
    if (i >= N_) return;
    int r = i % HW_;
    int y = r / W_;
    int y0 = max(y - RM, 0), y1 = min(y + RM, H_ - 1);
    float v = 3.0e38f;
    for (int yc = y0; yc <= y1; ++yc) v = fminf(v, in[i + (yc - y) * W_]);
    out[i] = v;
}

// ---------------- horizontal box-sum via LDS prefix scan ----------------
// One block per image row. Rows staged in LDS with async B128 DMA; inclusive
// scan S (WMMA matrix-pipe scan); hsum[x] = S[min(x+R,W-1)] - S[x-R-1].
__global__ __launch_bounds__(256) void k_hpass4(const float* __restrict__ I,
                                                const float* __restrict__ p,
                                                float* __restrict__ oI,
                                                float* __restrict__ op,
                                                float* __restrict__ oIp,
                                                float* __restrict__ oII) {
    __shared__ float rI[W_], rp[W_], S[W_], part[256], tt[16];
    const int t = threadIdx.x;
    const size_t row = (size_t)blockIdx.x * W_;
    // async stage: lane t copies floats [4t, 4t+4) of both rows into LDS
    async_g2l_b128((LDS_AS float*)&rI[4 * t], &I[row + 4 * t]);
    async_g2l_b128((LDS_AS float*)&rp[4 * t], &p[row + 4 * t]);
    wait_async0();
    __syncthreads();
    const int xb = 4 * t;
    for (int c = 0; c < 4; ++c) {
        float v[4];
        for (int j = 0; j < 4; ++j) {
            float a = rI[xb + j], b = rp[xb + j];
            v[j] = (c == 0) ? a : (c == 1) ? b : (c == 2) ? a * b : a * a;
        }
        v[1] += v[0]; v[2] += v[1]; v[3] += v[2];
        part[t] = v[3];
        scan256(part, tt, t);
        float base = (t > 0) ? part[t - 1] : 0.0f;
        for (int j = 0; j < 4; ++j) S[xb + j] = v[j] + base;
        __syncthreads();
        float* o = (c == 0) ? oI : (c == 1) ? op : (c == 2) ? oIp : oII;
        for (int j = 0; j < 4; ++j) {
            int xx = t + 256 * j;
            int hi = min(xx + RG, W_ - 1);
            int lo = xx - RG - 1;
            o[row + xx] = S[hi] - (lo >= 0 ? S[lo] : 0.0f);
        }
        __syncthreads();
    }
}

__global__ __launch_bounds__(256) void k_hpass2(const float* __restrict__ A,
                                                const float* __restrict__ Bm,
                                                float* __restrict__ oA,
                                                float* __restrict__ oB) {
    __shared__ float rA[W_], rB[W_], S[W_], part[256], tt[16];
    const int t = threadIdx.x;
    const size_t row = (size_t)blockIdx.x * W_;
    async_g2l_b128((LDS_AS float*)&rA[4 * t], &A[row + 4 * t]);
    async_g2l_b128((LDS_AS float*)&rB[4 * t], &Bm[row + 4 * t]);
    wait_async0();
    __syncthreads();
    const int xb = 4 * t;
    for (int c = 0; c < 2; ++c) {
        const float* src = (c == 0) ? rA : rB;
        float v[4];
        for (int j = 0; j < 4; ++j) v[j] = src[xb + j];
        v[1] += v[0]; v[2] += v[1]; v[3] += v[2];
        part[t] = v[3];
        scan256(part, tt, t);
        float base = (t > 0) ? part[t - 1] : 0.0f;
        for (int j = 0; j < 4; ++j) S[xb + j] = v[j] + base;
        __syncthreads();
        float* o = (c == 0) ? oA : oB;
        for (int j = 0; j < 4; ++j) {
            int xx = t + 256 * j;
            int hi = min(xx + RG, W_ - 1);
            int lo = xx - RG - 1;
            o[row + xx] = S[hi] - (lo >= 0 ? S[lo] : 0.0f);
        }
        __syncthreads();
    }
}

// ---------------- vertical box-sum (running sum down columns) ----------------
__global__ __launch_bounds__(256) void k_vpass4(const float* __restrict__ h0,
                                                const float* __restrict__ h1,
                                                const float* __restrict__ h2,
                                                const float* __restrict__ h3,
                                                float* __restrict__ oa,
                                                float* __restrict__ ob) {
    const int x = blockIdx.x * 256 + threadIdx.x;
    const int b = blockIdx.z;
    const int y0 = blockIdx.y * SEG;
    const int base = b * HW_ + x;
    float s0 = 0.f, s1 = 0.f, s2 = 0.f, s3 = 0.f;
    int ys = max(y0 - RG, 0);
    int ye = min(y0 + RG - 1, H_ - 1);  // rows [max(0,y0-81), y0+80]
    for (int yy = ys; yy <= ye; ++yy) {
        int idx = base + yy * W_;
        s0 += h0[idx]; s1 += h1[idx]; s2 += h2[idx]; s3 += h3[idx];
    }
    const float inv = 1.0f / KK2;
    for (int y = y0; y < y0 + SEG; ++y) {
        int ya = y + RG;
        if (ya < H_) {
            int idx = base + ya * W_;
            s0 += h0[idx]; s1 += h1[idx]; s2 += h2[idx]; s3 += h3[idx];
        }
        float mI = s0 * inv, mp = s1 * inv, mIp = s2 * inv, mII = s3 * inv;
        float var = mII - mI * mI;
        float a = (mIp - mI * mp) / (var + EPSG);
        float bb = mp - a * mI;
        int oidx = base + y * W_;
        oa[oidx] = a;
        ob[oidx] = bb;
        int yr = y - RG;
        if (yr >= 0) {
            int idx = base + yr * W_;
            s0 -= h0[idx]; s1 -= h1[idx]; s2 -= h2[idx]; s3 -= h3[idx];
        }
    }
}

__global__ __launch_bounds__(256) void k_vpass2(const float* __restrict__ ha,
                                                const float* __restrict__ hb,
                                                const float* __restrict__ I,
                                                float* __restrict__ q) {
    const int x = blockIdx.x * 256 + threadIdx.x;
    const int b = blockIdx.z;
    const int y0 = blockIdx.y * SEG;
    const int base = b * HW_ + x;
    float s0 = 0.f, s1 = 0.f;
    int ys = max(y0 - RG, 0);
    int ye = min(y0 + RG - 1, H_ - 1);
    for (int yy = ys; yy <= ye; ++yy) {
        int idx = base + yy * W_;
        s0 += ha[idx]; s1 += hb[idx];
    }
    const float inv = 1.0f / KK2;
    for (int y = y0; y < y0 + SEG; ++y) {
        int ya = y + RG;
        if (ya < H_) {
            int idx = base + ya * W_;
            s0 += ha[idx]; s1 += hb[idx];
        }
        int oidx = base + y * W_;
        q[oidx] = (s0 * inv) * I[oidx] + (s1 * inv);
        int yr = y - RG;
        if (yr >= 0) {
            int idx = base + yr * W_;
            s0 -= ha[idx]; s1 -= hb[idx];
        }
    }
}

// ---------------- global min/max (encoded atomics; deterministic) ----------------
__global__ __launch_bounds__(256) void k_minmax(const float* __restrict__ q,
                                                unsigned* aux) {
    float lmn = 3.4e38f, lmx = -3.4e38f;
    for (int i = blockIdx.x * 256 + threadIdx.x; i < N_; i += gridDim.x * 256) {
        float v = q[i];
        lmn = fminf(lmn, v);
        lmx = fmaxf(lmx, v);
    }
    __shared__ float smn[256], smx[256];
    int t = threadIdx.x;
    smn[t] = lmn; smx[t] = lmx;
    __syncthreads();
    for (int o = 128; o; o >>= 1) {
        if (t < o) { smn[t] = fminf(smn[t], smn[t + o]); smx[t] = fmaxf(smx[t], smx[t + o]); }
        __syncthreads();
    }
    if (t == 0) {
        atomicMin(&aux[0], fenc(smn[0]));
        atomicMax(&aux[1], fenc(smx[0]));
    }
}

// ---------------- histogram (LDS-privatized, integer atomics) ----------------
__global__ __launch_bounds__(256) void k_hist(const float* __restrict__ q,
                                              unsigned* aux) {
    __shared__ unsigned lh[BINS_];
    unsigned* gh = aux + 16;
    for (int i = threadIdx.x; i < BINS_; i += 256) lh[i] = 0u;
    __syncthreads();
    const float mn = fdec(aux[0]);
    const float mx = fdec(aux[1]);
    const float invr = 1.0f / (mx - mn);
    for (int i = blockIdx.x * 256 + threadIdx.x; i < N_; i += gridDim.x * 256) {
        int idx = (int)(((q[i] - mn) * invr) * (float)BINS_);
        idx = min(max(idx, 0), BINS_ - 1);
        atomicAdd(&lh[idx], 1u);
    }
    __syncthreads();
    for (int i = threadIdx.x; i < BINS_; i += 256)
        if (lh[i]) atomicAdd(&gh[i], lh[i]);
}

// ---------------- cumsum / threshold (tiny, serial) ----------------
__global__ void k_thresh(unsigned* aux) {
    if (threadIdx.x != 0 || blockIdx.x != 0) return;
    const unsigned* gh = aux + 16;
    float cum = 0.0f;
    int lmax = -1;
    for (int i = 0; i < BINS_; ++i) {
        cum += (float)gh[i];
        if (cum / (float)N_ <= 0.999f) lmax = i;
    }
    int idx = (lmax < 0) ? (BINS_ - 1) : lmax;  // jnp wraps -1 -> last bin
    ((float*)aux)[2] = (float)gh[idx];
}

// ---------------- per-image means (two-stage, deterministic) ----------------
__global__ __launch_bounds__(256) void k_fb(const float* __restrict__ x,
                                            float* __restrict__ partials) {
    const int b = blockIdx.y;
    const int blk = blockIdx.x;
    const int CH = 3 * HW_ / FBBLK;  // 24576
    const size_t base = (size_t)b * 3 * HW_ + (size_t)blk * CH;
    float s = 0.0f;
    for (int i = threadIdx.x; i < CH; i += 256) s += x[base + i];
    __shared__ float sm[256];
    sm[threadIdx.x] = s;
    __syncthreads();
    for (int o = 128; o; o >>= 1) {
        if (threadIdx.x < o) sm[threadIdx.x] += sm[threadIdx.x + o];
        __syncthreads();
    }
    if (threadIdx.x == 0) partials[b * FBBLK + blk] = sm[0];
}

__global__ void k_fb2(const float* __restrict__ partials, unsigned* aux) {
    __shared__ float means[B_];
    int t = threadIdx.x;
    if (t < B_) {
        float s = 0.0f;
        for (int i = 0; i < FBBLK; ++i) s += partials[t * FBBLK + i];
        means[t] = s * 255.0f / (float)(3 * HW_);
    }
    __syncthreads();
    if (t == 0) {
        float m = means[0];
        for (int i = 1; i < B_; ++i) m = fmaxf(m, means[i]);
        ((float*)aux)[6] = m;
    }
}

// ---------------- masked max of channel-mean (A candidate) ----------------
__global__ __launch_bounds__(256) void k_acand(const float* __restrict__ x,
                                               const float* __restrict__ q,
                                               unsigned* aux) {
    int i = blockIdx.x * 256 + threadIdx.x;
    if (i >= N_) return;
    float thresh = ((const float*)aux)[2];
    if (q[i] >= thresh) {
        int b = i / HW_, r = i - b * HW_;
        const float* p = x + (size_t)b * 3 * HW_ + r;
        float mm = (p[0] + p[HW_] + p[2 * HW_]) * (255.0f / 3.0f);
        atomicMax(&aux[5], fenc(mm));
        atomicOr(&aux[4], 1u);
    }
}

__global__ void k_afinal(unsigned* aux) {
    if (threadIdx.x != 0 || blockIdx.x != 0) return;
    float A = aux[4] ? fdec(aux[5]) : ((float*)aux)[6];
    ((float*)aux)[3] = A;
}

// ---------------- final correction ----------------
__global__ __launch_bounds__(256) void k_final(const float* __restrict__ x,
                                               const float* __restrict__ q,
                                               const unsigned* __restrict__ aux,
                                               float* __restrict__ out) {
    int i = blockIdx.x * 256 + threadIdx.x;
    if (i >= N_) return;
    const float A = ((const float*)aux)[3];
    float v1c = fminf(q[i] * 0.95f, 0.8f);
    float denom = 1.0f - v1c / A;
    int b = i / HW_, r = i - b * HW_;
    size_t o = (size_t)b * 3 * HW_ + r;
    for (int c = 0; c < 3; ++c) {
        float m = x[o + (size_t)c * HW_] * 255.0f;
        float y = (m - v1c) / denom;
        out[o + (size_t)c * HW_] = fminf(fmaxf(y, 0.0f), 1.0f);
    }
}

extern "C" void kernel_launch(void* const* d_in, const int* in_sizes, int n_in,
                              void* d_out, int out_size, void* d_ws, size_t ws_size,
                              hipStream_t stream) {
    const float* x = (const float*)d_in[0];
    float* out = (float*)d_out;
    float* ws = (float*)d_ws;
    const size_t PN = (size_t)N_;
    float* P0 = ws + 0 * PN;  // dark channel I (guide)
    float* P1 = ws + 1 * PN;  // rowmin tmp / hsum_I / hsum_a
    float* P2 = ws + 2 * PN;  // p / a
    float* P3 = ws + 3 * PN;  // hsum_p / hsum_b
    float* P4 = ws + 4 * PN;  // hsum_Ip / q
    float* P5 = ws + 5 * PN;  // hsum_II
    float* P6 = ws + 6 * PN;  // b
    unsigned* aux = (unsigned*)(ws + 7 * PN);
    float* partials = (float*)(aux + AUX_WORDS);

    const int NB = N_ / 256;  // 24576
    dim3 vgrid(W_ / 256, NSEG, B_);

    k_init<<<1, 256, 0, stream>>>(aux);
    k_dark<<<NB, 256, 0, stream>>>(x, P0);
    k_rowmin<<<NB, 256, 0, stream>>>(P0, P1);
    k_colmin<<<NB, 256, 0, stream>>>(P1, P2);                       // P2 = p
    k_hpass4<<<B_ * H_, 256, 0, stream>>>(P0, P2, P1, P3, P4, P5);  // h-sums
    k_vpass4<<<vgrid, 256, 0, stream>>>(P1, P3, P4, P5, P2, P6);    // P2=a, P6=b
    k_hpass2<<<B_ * H_, 256, 0, stream>>>(P2, P6, P1, P3);          // h-sums of a,b
    k_vpass2<<<vgrid, 256, 0, stream>>>(P1, P3, P0, P4);            // P4 = q (filtered V1)
    k_minmax<<<2048, 256, 0, stream>>>(P4, aux);
    k_hist<<<2048, 256, 0, stream>>>(P4, aux);
    k_thresh<<<1, 32, 0, stream>>>(aux);
    k_fb<<<dim3(FBBLK, B_), 256, 0, stream>>>(x, partials);
    k_fb2<<<1, 64, 0, stream>>>(partials, aux);
    k_acand<<<NB, 256, 0, stream>>>(x, P4, aux);
    k_afinal<<<1, 32, 0, stream>>>(aux);
    k_final<<<NB, 256, 0, stream>>>(x, P4, aux, out);
}